// GNN_v1_33500744908950
// MI455X (gfx1250) — compile-verified
//
#include <hip/hip_runtime.h>

typedef float v2f __attribute__((ext_vector_type(2)));
typedef float v8f __attribute__((ext_vector_type(8)));

#define HIDDEN   256
#define CONV_OUT 128

// ---------------------------------------------------------------------------
// 1) deg[i] = 1.0 (self-loop weight)
// ---------------------------------------------------------------------------
__global__ __launch_bounds__(256) void gcn_deg_init(float* __restrict__ deg, int n) {
    int i = blockIdx.x * blockDim.x + threadIdx.x;
    if (i < n) deg[i] = 1.0f;
}

// ---------------------------------------------------------------------------
// 2) deg[dst[e]] += w[e]
// ---------------------------------------------------------------------------
__global__ __launch_bounds__(256) void gcn_deg_acc(const long long* __restrict__ ei,
                                                   const float* __restrict__ w,
                                                   float* __restrict__ deg, int E) {
    int e = blockIdx.x * blockDim.x + threadIdx.x;
    if (e < E) {
        long long d = ei[(size_t)E + e];
        unsafeAtomicAdd(&deg[d], w[e]);
    }
}

// ---------------------------------------------------------------------------
// 3) dinv[i] = deg>0 ? rsqrt(deg) : 0   (in place)
// ---------------------------------------------------------------------------
__global__ __launch_bounds__(256) void gcn_dinv(float* __restrict__ deg, int n) {
    int i = blockIdx.x * blockDim.x + threadIdx.x;
    if (i < n) {
        float d = deg[i];
        deg[i] = (d > 0.0f) ? rsqrtf(d) : 0.0f;
    }
}

// ---------------------------------------------------------------------------
// 4) h = x @ W1 via V_WMMA_F32_16X16X4_F32.
//    One wave -> 16x32 output (two 16x16 accumulators sharing the A fragment).
//    Block = 8 waves: waves 0-3 cover rows [0,16) x cols 0..127,
//                     waves 4-7 cover rows [16,32). Block = 32-row stripe.
//    A-frag (16x4 f32): lanes 0-15 hold K=0,1 ; lanes 16-31 hold K=2,3.
//    B-frag (4x16 f32): lanes 0-15 hold rows K=0,1 ; lanes 16-31 rows K=2,3.
//    D (16x16 f32): VGPR r -> M=r (lanes 0-15) / M=r+8 (lanes 16-31).
// ---------------------------------------------------------------------------
__global__ __launch_bounds__(256) void gcn_gemm_wmma(const float* __restrict__ x,
                                                     const float* __restrict__ W1,
                                                     float* __restrict__ h, int n) {
    const int lane  = threadIdx.x & 31;
    const int wave  = threadIdx.x >> 5;        // 0..7
    const int r16   = lane & 15;
    const int khalf = lane >> 4;               // 0 -> K{0,1}, 1 -> K{2,3}

    const int rowbase = blockIdx.x * 32 + (wave >> 2) * 16;  // 16-row sub-stripe
    int row  = rowbase + r16;
    int rowc = row < n ? row : (n - 1);        // clamp tail loads (EXEC stays full)
    const int col0 = (wave & 3) * 32 + r16;    // first of two adjacent col tiles
    const int col1 = col0 + 16;

    const float* __restrict__ xr  = x + (size_t)rowc * HIDDEN + khalf * 2;
    const float* __restrict__ wp0 = W1 + (size_t)(khalf * 2) * CONV_OUT + col0;
    const float* __restrict__ wp1 = W1 + (size_t)(khalf * 2) * CONV_OUT + col1;

    v8f acc0 = {}, acc1 = {};
#pragma unroll 8
    for (int k = 0; k < HIDDEN; k += 4) {
        if ((k & 63) == 0) __builtin_prefetch(xr + k + 64, 0, 3);  // WGP-scope prefetch
        v2f a = *(const v2f*)(xr + k);                 // x[row][k+khalf*2 .. +1]
        v2f b0, b1;
        b0[0] = wp0[(size_t)k * CONV_OUT];             // W1[k+khalf*2 ][col0]
        b0[1] = wp0[(size_t)k * CONV_OUT + CONV_OUT];  // W1[k+khalf*2+1][col0]
        b1[0] = wp1[(size_t)k * CONV_OUT];
        b1[1] = wp1[(size_t)k * CONV_OUT + CONV_OUT];
        acc0 = __builtin_amdgcn_wmma_f32_16x16x4_f32(
            false, a, false, b0, (short)0, acc0, false, false);
        acc1 = __builtin_amdgcn_wmma_f32_16x16x4_f32(
            false, a, false, b1, (short)0, acc1, false, false);
    }

    const int mbase = rowbase + khalf * 8;
    if (rowbase + 15 < n) {
        // fast path: whole 16-row sub-stripe in range -> unconditional stores
#pragma unroll
        for (int r = 0; r < 8; ++r) {
            size_t ro = (size_t)(mbase + r) * CONV_OUT;
            h[ro + col0] = acc0[r];
            h[ro + col1] = acc1[r];
        }
    } else {
#pragma unroll
        for (int r = 0; r < 8; ++r) {
            int m = mbase + r;
            if (m < n) {
                size_t ro = (size_t)m * CONV_OUT;
                h[ro + col0] = acc0[r];
                h[ro + col1] = acc1[r];
            }
        }
    }
}

// ---------------------------------------------------------------------------
// 5) agg[i][f] = b1[f] + dinv[i]^2 * h[i][f]   (bias + self-loop term)
// ---------------------------------------------------------------------------
__global__ __launch_bounds__(256) void gcn_agg_init(const float* __restrict__ h,
                                                    const float* __restrict__ dinv,
                                                    const float* __restrict__ b1,
                                                    float* __restrict__ agg, int n) {
    size_t i4 = ((size_t)blockIdx.x * blockDim.x + threadIdx.x) * 4;
    size_t total = (size_t)n * CONV_OUT;
    if (i4 >= total) return;
    int node = (int)(i4 >> 7);
    int f    = (int)(i4 & 127);
    float di = dinv[node];
    float s  = di * di;
    float4 hv = *(const float4*)(h + i4);
    float4 bv = *(const float4*)(b1 + f);
    float4 o;
    o.x = bv.x + s * hv.x;
    o.y = bv.y + s * hv.y;
    o.z = bv.z + s * hv.z;
    o.w = bv.w + s * hv.w;
    *(float4*)(agg + i4) = o;
}

// ---------------------------------------------------------------------------
// 6) edge scatter: one wave32 per edge, lane -> 4 features.
//    agg[dst] += dinv[src]*w*dinv[dst] * h[src]
// ---------------------------------------------------------------------------
__global__ __launch_bounds__(256) void gcn_scatter(const long long* __restrict__ ei,
                                                   const float* __restrict__ w,
                                                   const float* __restrict__ dinv,
                                                   const float* __restrict__ h,
                                                   float* __restrict__ agg, int E) {
    int wave = (int)(((size_t)blockIdx.x * blockDim.x + threadIdx.x) >> 5);
    int lane = threadIdx.x & 31;
    if (wave >= E) return;
    long long s = ei[wave];
    long long d = ei[(size_t)E + wave];
    float norm = dinv[s] * w[wave] * dinv[d];
    float4 hv = *(const float4*)(h + (size_t)s * CONV_OUT + lane * 4);
    float* ap = agg + (size_t)d * CONV_OUT + lane * 4;
    unsafeAtomicAdd(ap + 0, norm * hv.x);
    unsafeAtomicAdd(ap + 1, norm * hv.y);
    unsafeAtomicAdd(ap + 2, norm * hv.z);
    unsafeAtomicAdd(ap + 3, norm * hv.w);
}

// ---------------------------------------------------------------------------
// 7) out = agg @ W2 + b2   (128 -> 2)
// ---------------------------------------------------------------------------
__global__ __launch_bounds__(256) void gcn_head(const float* __restrict__ agg,
                                                const float* __restrict__ W2,
                                                const float* __restrict__ b2,
                                                float* __restrict__ out, int n) {
    int node = blockIdx.x * blockDim.x + threadIdx.x;
    if (node >= n) return;
    float a0 = b2[0], a1 = b2[1];
    const float* a = agg + (size_t)node * CONV_OUT;
#pragma unroll 8
    for (int k = 0; k < CONV_OUT; ++k) {
        float av = a[k];
        a0 = fmaf(av, W2[2 * k + 0], a0);
        a1 = fmaf(av, W2[2 * k + 1], a1);
    }
    out[2 * node + 0] = a0;
    out[2 * node + 1] = a1;
}

// ---------------------------------------------------------------------------
extern "C" void kernel_launch(void* const* d_in, const int* in_sizes, int n_in,
                              void* d_out, int out_size, void* d_ws, size_t ws_size,
                              hipStream_t stream) {
    const float*     x  = (const float*)d_in[0];
    const long long* ei = (const long long*)d_in[1];   // int64 edge_index [2, E]
    const float*     ew = (const float*)d_in[2];
    const float*     W1 = (const float*)d_in[3];
    const float*     b1 = (const float*)d_in[4];
    const float*     W2 = (const float*)d_in[5];
    const float*     b2 = (const float*)d_in[6];
    float*           out = (float*)d_out;

    const int N = in_sizes[0] / HIDDEN;   // 100000
    const int E = in_sizes[2];            // 1600000

    // workspace layout: dinv[N] | h[N*128] | agg[N*128]
    float* dinv = (float*)d_ws;
    float* h    = dinv + N;
    float* agg  = h + (size_t)N * CONV_OUT;

    const int T = 256;
    gcn_deg_init<<<(N + T - 1) / T, T, 0, stream>>>(dinv, N);
    gcn_deg_acc <<<(E + T - 1) / T, T, 0, stream>>>(ei, ew, dinv, E);
    gcn_dinv    <<<(N + T - 1) / T, T, 0, stream>>>(dinv, N);

    gcn_gemm_wmma<<<(N + 31) / 32, T, 0, stream>>>(x, W1, h, N);

    size_t total = (size_t)N * CONV_OUT;
    gcn_agg_init<<<(unsigned)((total / 4 + T - 1) / T), T, 0, stream>>>(h, dinv, b1, agg, N);

    gcn_scatter<<<(E + 7) / 8, T, 0, stream>>>(ei, ew, dinv, h, agg, E);

    gcn_head<<<(N + T - 1) / T, T, 0, stream>>>(agg, W2, b2, out, N);
}